// FixedEntropyHardNegativeLoss_1752346657491
// MI455X (gfx1250) — compile-verified
//
#include <hip/hip_runtime.h>

typedef float v2f __attribute__((ext_vector_type(2)));
typedef float v4f __attribute__((ext_vector_type(4)));
typedef float v8f __attribute__((ext_vector_type(8)));

#define BPTS 128
#define DIM  128
#define NBANK 1280000
#define KSEL 4096
#define CAND_CAP 16384
#define CHUNKS 10              // per-row chunks for hist/gather: 128000 elems each
#define EPSF 1e-7f
#define TARGET_ENT 5.0f

__device__ __forceinline__ unsigned int fkey(float f) {
    unsigned int u = __float_as_uint(f);
    return (u & 0x80000000u) ? ~u : (u | 0x80000000u);
}
__device__ __forceinline__ float unkey(unsigned int k) {
    unsigned int u = (k & 0x80000000u) ? (k & 0x7FFFFFFFu) : ~k;
    return __uint_as_float(u);
}

// ---------------------------------------------------------------- zero scratch
__global__ void zero_u32(unsigned int* __restrict__ p, long long n) {
    long long i = (long long)blockIdx.x * blockDim.x + threadIdx.x;
    long long stride = (long long)gridDim.x * blockDim.x;
    for (; i < n; i += stride) p[i] = 0u;
}

// ---------------------------------------------------------------- normalize points
__global__ __launch_bounds__(128)
void norm_kernel(const float* __restrict__ points, float* __restrict__ normP) {
    __shared__ float red[128];
    int row = blockIdx.x, tid = threadIdx.x;
    float x = points[row * DIM + tid];
    red[tid] = x * x;
    __syncthreads();
    for (int off = 64; off > 0; off >>= 1) {
        if (tid < off) red[tid] += red[tid + off];
        __syncthreads();
    }
    float nrm = sqrtf(red[0]);
    normP[row * DIM + tid] = x / nrm;
}

// ---------------------------------------------------------------- WMMA GEMM: sims = P_norm @ bank^T
__global__ __launch_bounds__(256)
void gemm_wmma(const float* __restrict__ normP, const float* __restrict__ bank,
               float* __restrict__ sims) {
    __shared__ float ldsP[BPTS * DIM];                 // 64 KB, all normalized points
    for (int i = threadIdx.x; i < BPTS * DIM; i += 256) ldsP[i] = normP[i];
    __syncthreads();

    const int lane  = threadIdx.x & 31;
    const int wave  = threadIdx.x >> 5;
    const int nrow  = lane & 15;        // N index within 16-wide tile (and A-row offset)
    const int khalf = lane >> 4;        // 0/1 -> K offset 0/2 per ISA 16x4 f32 layout
    const long long n0 = (long long)blockIdx.x * 128 + wave * 16;

    const float* bptr = bank + (n0 + nrow) * DIM + khalf * 2;

    v8f acc[8];
    v8f zero = {0.f, 0.f, 0.f, 0.f, 0.f, 0.f, 0.f, 0.f};
#pragma unroll
    for (int m = 0; m < 8; ++m) acc[m] = zero;

    for (int kk = 0; kk < DIM; kk += 4) {
        v2f bfrag = __builtin_nontemporal_load((const v2f*)(bptr + kk));   // bank streamed once
#pragma unroll
        for (int m = 0; m < 8; ++m) {
            v2f afrag = *(const v2f*)&ldsP[(m * 16 + nrow) * DIM + kk + khalf * 2];
            acc[m] = __builtin_amdgcn_wmma_f32_16x16x4_f32(
                false, afrag, false, bfrag, (short)0, acc[m], false, false);
        }
    }

    const long long colBase = n0 + nrow;
#pragma unroll
    for (int m = 0; m < 8; ++m) {
#pragma unroll
        for (int v = 0; v < 8; ++v) {
            int row = m * 16 + v + 8 * khalf;          // C layout: VGPR v, lane-half selects M+8
            __builtin_nontemporal_store(acc[m][v], &sims[(long long)row * NBANK + colBase]);
        }
    }
}

// ---------------------------------------------------------------- per-row 12-bit key histogram
// sims rows start at byte offset ≡ 4 (mod 16): peel 3 head + 1 tail, vec4 body aligned.
#define CHUNK_ELEMS (NBANK / CHUNKS)            // 128000
#define CHUNK_V4    ((CHUNK_ELEMS - 4) / 4)     // 31999 aligned vec4 after 3-elem peel

__global__ __launch_bounds__(256)
void hist_kernel(const float* __restrict__ sims, unsigned int* __restrict__ hist) {
    __shared__ unsigned int h[4096];
    for (int i = threadIdx.x; i < 4096; i += 256) h[i] = 0u;
    __syncthreads();
    const int row = blockIdx.y;
    const float* base = sims + (long long)row * NBANK + (long long)blockIdx.x * CHUNK_ELEMS;
    if (threadIdx.x < 3)
        atomicAdd(&h[fkey(base[threadIdx.x]) >> 20], 1u);
    const v4f* v4 = (const v4f*)(base + 3);
    for (int i = threadIdx.x; i < CHUNK_V4; i += 256) {
        v4f v = __builtin_nontemporal_load(&v4[i]);
        atomicAdd(&h[fkey(v[0]) >> 20], 1u);
        atomicAdd(&h[fkey(v[1]) >> 20], 1u);
        atomicAdd(&h[fkey(v[2]) >> 20], 1u);
        atomicAdd(&h[fkey(v[3]) >> 20], 1u);
    }
    if (threadIdx.x == 0)
        atomicAdd(&h[fkey(base[CHUNK_ELEMS - 1]) >> 20], 1u);
    __syncthreads();
    unsigned int* grow = hist + (long long)row * 4096;
    for (int i = threadIdx.x; i < 4096; i += 256) {
        unsigned int c = h[i];
        if (c) atomicAdd(&grow[i], c);
    }
}

// ---------------------------------------------------------------- threshold bin per row (parallel scan)
__global__ __launch_bounds__(256)
void find_bin(const unsigned int* __restrict__ hist, int* __restrict__ bininfo) {
    __shared__ unsigned int h[4096];
    __shared__ unsigned int seg[256];
    const int row = blockIdx.x, tid = threadIdx.x;
    const unsigned int* grow = hist + (long long)row * 4096;
    for (int i = tid; i < 4096; i += 256) h[i] = grow[i];
    __syncthreads();
    const int base = tid * 16;
    unsigned int s = 0;
    for (int i = 0; i < 16; ++i) s += h[base + i];
    seg[tid] = s;
    __syncthreads();
    // inclusive suffix scan over segment sums (Hillis-Steele)
    for (int off = 1; off < 256; off <<= 1) {
        unsigned int v = seg[tid] + ((tid + off < 256) ? seg[tid + off] : 0u);
        __syncthreads();
        seg[tid] = v;
        __syncthreads();
    }
    unsigned int S = (tid + 1 < 256) ? seg[tid + 1] : 0u;   // sum of bins >= (tid+1)*16
    for (int b = base + 15; b >= base; --b) {
        S += h[b];
        unsigned int above = S - h[b];
        if (S >= (unsigned)KSEL && above < (unsigned)KSEL) {
            bininfo[row * 2 + 0] = b;
            bininfo[row * 2 + 1] = (int)above;   // exactly one bin matches per row
        }
    }
}

// ---------------------------------------------------------------- gather top values + threshold-bin candidates
__device__ __forceinline__ void gather_one(float v, int bstar, int row,
                                           int* aboveCnt, int* candCnt,
                                           float* topbuf, float* candbuf) {
    int b = (int)(fkey(v) >> 20);
    if (b > bstar) {
        int s = atomicAdd(&aboveCnt[row], 1);
        if (s < KSEL) topbuf[(long long)row * KSEL + s] = v;
    } else if (b == bstar) {
        int s = atomicAdd(&candCnt[row], 1);
        if (s < CAND_CAP) candbuf[(long long)row * CAND_CAP + s] = v;
    }
}

__global__ __launch_bounds__(256)
void gather_kernel(const float* __restrict__ sims, const int* __restrict__ bininfo,
                   int* __restrict__ aboveCnt, int* __restrict__ candCnt,
                   float* __restrict__ topbuf, float* __restrict__ candbuf) {
    const int row = blockIdx.y;
    const int bstar = bininfo[row * 2];
    const float* base = sims + (long long)row * NBANK + (long long)blockIdx.x * CHUNK_ELEMS;
    if (threadIdx.x < 3)
        gather_one(base[threadIdx.x], bstar, row, aboveCnt, candCnt, topbuf, candbuf);
    const v4f* v4 = (const v4f*)(base + 3);
    for (int i = threadIdx.x; i < CHUNK_V4; i += 256) {
        v4f v = __builtin_nontemporal_load(&v4[i]);
        gather_one(v[0], bstar, row, aboveCnt, candCnt, topbuf, candbuf);
        gather_one(v[1], bstar, row, aboveCnt, candCnt, topbuf, candbuf);
        gather_one(v[2], bstar, row, aboveCnt, candCnt, topbuf, candbuf);
        gather_one(v[3], bstar, row, aboveCnt, candCnt, topbuf, candbuf);
    }
    if (threadIdx.x == 0)
        gather_one(base[CHUNK_ELEMS - 1], bstar, row, aboveCnt, candCnt, topbuf, candbuf);
}

// ---------------------------------------------------------------- per-row: radix-select candidates, sort top-4096
__device__ __forceinline__ int block_reduce_int1024(int v, int* red) {
    int tid = threadIdx.x;
    red[tid] = v;
    __syncthreads();
    for (int off = 512; off > 0; off >>= 1) {
        if (tid < off) red[tid] += red[tid + off];
        __syncthreads();
    }
    int r = red[0];
    __syncthreads();
    return r;
}

// key of k-th largest (1-based) element of arr[0..n) -- pure counting, order-independent
__device__ unsigned int radix_kth_key(const float* arr, int n, int k, int* red) {
    unsigned int prefix = 0u;
    int need = k;
    for (int bit = 31; bit >= 0; --bit) {
        unsigned int test = 1u << bit;
        unsigned int highmask = (bit == 31) ? 0u : (~0u << (bit + 1));
        int c = 0;
        for (int i = threadIdx.x; i < n; i += 1024) {
            unsigned int key = fkey(arr[i]);
            if ((key & highmask) == prefix && (key & test)) c++;
        }
        c = block_reduce_int1024(c, red);
        if (c >= need) prefix |= test;
        else need -= c;
    }
    return prefix;
}

__global__ __launch_bounds__(1024)
void select_kernel(const int* __restrict__ aboveCnt, const int* __restrict__ candCnt,
                   float* __restrict__ topbuf, const float* __restrict__ candbuf) {
    __shared__ float cand[CAND_CAP];   // 64 KB
    __shared__ float top[KSEL];        // 16 KB
    __shared__ int red[1024];
    __shared__ int ctr;
    const int row = blockIdx.x, tid = threadIdx.x;
    int cab = aboveCnt[row]; if (cab > KSEL) cab = KSEL;
    const int krem = KSEL - cab;
    int nc = candCnt[row]; if (nc > CAND_CAP) nc = CAND_CAP;
    for (int i = tid; i < cab; i += 1024) top[i] = topbuf[(long long)row * KSEL + i];
    for (int i = tid; i < nc; i += 1024) cand[i] = candbuf[(long long)row * CAND_CAP + i];
    if (tid == 0) ctr = 0;
    __syncthreads();

    const int ksel = (krem < nc) ? krem : nc;
    unsigned int K2 = (ksel > 0) ? radix_kth_key(cand, nc, ksel, red) : 0u;
    const float tieval = unkey(K2);
    for (int i = tid; i < nc; i += 1024) {
        if (fkey(cand[i]) > K2) {
            int s0 = atomicAdd(&ctr, 1);
            if (s0 < krem) top[cab + s0] = cand[i];
        }
    }
    __syncthreads();
    int c2 = ctr; if (c2 > krem) c2 = krem;
    for (int i = c2 + tid; i < krem; i += 1024) top[cab + i] = tieval;
    __syncthreads();

    // bitonic sort descending over 4096 -> deterministic order, direct rank-p addressing
    for (int k = 2; k <= KSEL; k <<= 1) {
        for (int j = k >> 1; j > 0; j >>= 1) {
            for (int i = tid; i < KSEL; i += 1024) {
                int ixj = i ^ j;
                if (ixj > i) {
                    float a = top[i], b = top[ixj];
                    bool descend = ((i & k) == 0);
                    if (descend ? (a < b) : (a > b)) { top[i] = b; top[ixj] = a; }
                }
            }
            __syncthreads();
        }
    }
    for (int i = tid; i < KSEL; i += 1024) topbuf[(long long)row * KSEL + i] = top[i];
}

// ---------------------------------------------------------------- per-row entropy binary search + loss pieces
__device__ __forceinline__ float block_reduce256(float v, float* red) {
    int tid = threadIdx.x;
    red[tid] = v;
    __syncthreads();
    for (int off = 128; off > 0; off >>= 1) {
        if (tid < off) red[tid] += red[tid + off];
        __syncthreads();
    }
    float r = red[0];
    __syncthreads();
    return r;
}

__global__ __launch_bounds__(256)
void entropy_kernel(const float* __restrict__ topbuf, const int* __restrict__ pidx,
                    const float* __restrict__ sims, float* __restrict__ centers,
                    float* __restrict__ entv, float* __restrict__ rloss) {
    __shared__ float s[KSEL];
    __shared__ float F[KSEL];
    __shared__ float red[256];
    const int row = blockIdx.x;
    const int tid = threadIdx.x;
    for (int i = tid; i < KSEL; i += 256) s[i] = topbuf[(long long)row * KSEL + i];
    const int p = pidx[row];
    __syncthreads();

    float scale = 2.5f;
    float center = 5.0f;
    float entropy = 0.0f;
    for (int it = 0; it < 14; ++it) {
        float t = center;
        float ls = 0.0f;
        for (int i = tid; i < KSEL; i += 256) {
            float f = (i == p) ? 0.0f : expf(s[i] / t);
            F[i] = f;
            ls += f;
        }
        float S = block_reduce256(ls, red);
        float le = 0.0f;
        for (int i = tid; i < KSEL; i += 256) {
            float r = F[i] / S;
            le -= r * logf(r + EPSF);
        }
        entropy = block_reduce256(le, red);
        if (it < 13) {
            float ind = (entropy < TARGET_ENT) ? 1.0f : -1.0f;   // NaN entropy -> -1, matches jnp
            center = center + scale * ind;
            scale *= 0.5f;
        }
    }

    const float inv_c = 1.0f / center;
    float ld = 0.0f;
    for (int i = tid; i < KSEL; i += 256) ld += expf(s[i] * inv_c - inv_c);
    float den = block_reduce256(ld, red);
    if (tid == 0) {
        centers[row] = center;
        entv[row]    = entropy;
        float pos = sims[(long long)row * NBANK + p];
        float num = expf(pos * inv_c - inv_c);
        rloss[row] = -logf(num / den + EPSF);
    }
}

// ---------------------------------------------------------------- final scalar reductions
__global__ void finalize_kernel(const float* __restrict__ centers, const float* __restrict__ entv,
                                const float* __restrict__ rloss, float* __restrict__ out_loss,
                                float* __restrict__ out_mc, float* __restrict__ out_me) {
    __shared__ float rc[128], re[128], rl[128];
    int t = threadIdx.x;
    rc[t] = centers[t]; re[t] = entv[t]; rl[t] = rloss[t];
    __syncthreads();
    for (int off = 64; off > 0; off >>= 1) {
        if (t < off) { rc[t] += rc[t + off]; re[t] += re[t + off]; rl[t] += rl[t + off]; }
        __syncthreads();
    }
    if (t == 0) {
        *out_loss = rl[0] * (1.0f / 128.0f);
        *out_mc   = rc[0] * (1.0f / 128.0f);
        *out_me   = re[0] * (1.0f / 128.0f);
    }
}

// ---------------------------------------------------------------- launch
extern "C" void kernel_launch(void* const* d_in, const int* in_sizes, int n_in,
                              void* d_out, int out_size, void* d_ws, size_t ws_size,
                              hipStream_t stream) {
    const float* points = (const float*)d_in[0];
    const int*   pidx   = (const int*)d_in[1];
    const float* bank   = (const float*)d_in[2];

    float* out = (float*)d_out;
    const long long SIMS = (long long)BPTS * NBANK;
    float* out_loss = out;                // tuple order: loss, similarities, mean_centers, mean_entropy
    float* sims     = out + 1;
    float* out_mc   = out + 1 + SIMS;
    float* out_me   = out + 2 + SIMS;

    unsigned int* ws = (unsigned int*)d_ws;     // word-addressed scratch layout
    float*        normP    = (float*)(ws + 0);           // 16384
    unsigned int* hist     = ws + 16384;                  // 524288
    int*          aboveCnt = (int*)(ws + 540672);         // 128
    int*          candCnt  = (int*)(ws + 540800);         // 128
    int*          bininfo  = (int*)(ws + 540928);         // 256
    float*        centers  = (float*)(ws + 541184);       // 128
    float*        entv     = (float*)(ws + 541312);       // 128
    float*        rlossb   = (float*)(ws + 541440);       // 128
    float*        topbuf   = (float*)(ws + 541568);       // 524288
    float*        candbuf  = (float*)(ws + 1065856);      // 2097152  (~12.1 MB total)

    zero_u32<<<512, 256, 0, stream>>>(hist, (long long)(524288 + 256)); // hist + both counters
    norm_kernel<<<BPTS, DIM, 0, stream>>>(points, normP);
    gemm_wmma<<<NBANK / 128, 256, 0, stream>>>(normP, bank, sims);
    hist_kernel<<<dim3(CHUNKS, BPTS), 256, 0, stream>>>(sims, hist);
    find_bin<<<BPTS, 256, 0, stream>>>(hist, bininfo);
    gather_kernel<<<dim3(CHUNKS, BPTS), 256, 0, stream>>>(sims, bininfo, aboveCnt, candCnt, topbuf, candbuf);
    select_kernel<<<BPTS, 1024, 0, stream>>>(aboveCnt, candCnt, topbuf, candbuf);
    entropy_kernel<<<BPTS, 256, 0, stream>>>(topbuf, pidx, sims, centers, entv, rlossb);
    finalize_kernel<<<1, BPTS, 0, stream>>>(centers, entv, rlossb, out_loss, out_mc, out_me);
}